// DCGRUCell_59880434041185
// MI455X (gfx1250) — compile-verified
//
#include <hip/hip_runtime.h>
#include <math.h>

// ---------------- problem constants ----------------
#define N_NODES   4096
#define B_BATCH   64
#define F_IN      66                    // INPUT_DIM(2) + NUM_UNITS(64)
#define NUM_MAT   5                     // 2*K + 1
#define FB        (F_IN * B_BATCH)      // 4224
#define KTOT      (F_IN * NUM_MAT)      // 330
#define NNZ_E     65536
#define R_ROWS    (B_BATCH * N_NODES)   // 262144 GEMM rows

static constexpr size_t NS  = (size_t)N_NODES * FB; // 17,301,504 floats per xs buffer
static constexpr int    NSi = (int)NS;              // fits in int32 (element offset)

typedef float v2f __attribute__((ext_vector_type(2)));
typedef float v8f __attribute__((ext_vector_type(8)));

// ---------------- x0 build: x0[n, f, b]; f<2 from inputs, else state ----------------
// state = hx (use_r==0) or r*hx (use_r==1, r = value[row*128 + u], already sigmoided)
__global__ void build_x0_kernel(const float* __restrict__ inputs,
                                const float* __restrict__ hx,
                                const float* __restrict__ value,
                                float* __restrict__ x0, int use_r)
{
    size_t idx = (size_t)blockIdx.x * blockDim.x + threadIdx.x;
    if (idx >= NS) return;
    int b = (int)(idx % B_BATCH);
    int f = (int)((idx / B_BATCH) % F_IN);
    int n = (int)(idx / FB);
    float val;
    if (f < 2) {
        val = inputs[(size_t)b * (N_NODES * 2) + n * 2 + f];
    } else {
        int u = f - 2;
        size_t row = (size_t)b * N_NODES + n;
        float h = hx[row * 64 + u];
        if (use_r) h *= value[row * 128 + u];
        val = h;
    }
    x0[idx] = val;
}

// ---------------- zero fill (float4) ----------------
__global__ void zero_kernel(float4* __restrict__ p, size_t nquads)
{
    size_t i = (size_t)blockIdx.x * blockDim.x + threadIdx.x;
    if (i < nquads) p[i] = make_float4(0.f, 0.f, 0.f, 0.f);
}

// ---------------- SpMM: y[r,:] += v * x[c,:] via f32 HW atomics ----------------
// grid: (NNZ, ceil(1056/256)); each thread: one edge, 4 consecutive columns
__global__ void spmm_kernel(const int* __restrict__ rows, const int* __restrict__ cols,
                            const float* __restrict__ vals,
                            const float* __restrict__ x, float* __restrict__ y)
{
    int e  = blockIdx.x;
    int jq = blockIdx.y * blockDim.x + threadIdx.x;
    if (jq >= FB / 4) return;
    int r = rows[e], c = cols[e];
    float v = vals[e];
    const float4* xr = (const float4*)(x + (size_t)c * FB);
    float4 xv = xr[jq];
    float* yp = y + (size_t)r * FB + jq * 4;
    unsafeAtomicAdd(yp + 0, v * xv.x);
    unsafeAtomicAdd(yp + 1, v * xv.y);
    unsafeAtomicAdd(yp + 2, v * xv.z);
    unsafeAtomicAdd(yp + 3, v * xv.w);
}

// ---------------- Chebyshev combine: t = 2*t - x_pp (in place) ----------------
__global__ void cheb_kernel(float* __restrict__ t, const float* __restrict__ xpp)
{
    size_t i = (size_t)blockIdx.x * blockDim.x + threadIdx.x;
    if (i < NS) t[i] = 2.f * t[i] - xpp[i];
}

// ---------------- WMMA GEMM: out[row, :] = act(xcat[row, :] @ W + bias) ----------------
// xcat[row, k] = xs[k%5][n, (k/5)*B + b], row = b*N + n.
// Tiling: WMMA M-rows are 16 consecutive *b* values at fixed n -> each A fragment
// load is 16 consecutive floats across lanes (one 64B segment, fully coalesced).
// Each wave: 2 b-tiles (32 rows) x 64 cols = 8 f32 accumulators, 82 branchless
// K-steps of V_WMMA_F32_16X16X4_F32 + 1 predicated tail (K = 330 = 82*4 + 2).
// A/B fragment layout (ISA 7.12.2, 32-bit 16x4): lanes 0-15 hold K=k0,k0+1;
// lanes 16-31 hold K=k0+2,k0+3; M (resp. N) = lane & 15.
template <int OC, bool FUSE_FINAL>
__global__ __launch_bounds__(256)
void gemm_wmma_kernel(const float* __restrict__ xs_base,   // 5 contiguous buffers, stride NS
                      const float* __restrict__ W,         // [330, OC]
                      const float* __restrict__ bias,      // [OC]
                      const float* __restrict__ value,     // [R,128] (u gate), FUSE only
                      const float* __restrict__ hx,        // [R,64], FUSE only
                      float* __restrict__ out)             // [R, OC] or [R, 64]
{
    constexpr int colGroups = OC / 64;

    int waveId = blockIdx.x * (blockDim.x >> 5) + (threadIdx.x >> 5);
    int colGrp = waveId % colGroups;
    int tmp    = waveId / colGroups;
    int bGrp   = tmp & 1;            // two 32-row b-groups cover B = 64
    int n      = tmp >> 1;

    int lane  = threadIdx.x & 31;
    int laneM = lane & 15;
    int hi    = lane >> 4;           // 0: K=k0,k0+1   1: K=k0+2,k0+3
    int koff  = hi * 2;

    int bBase = bGrp * 32;
    // per-lane A base: row b = bBase + 16*t + laneM; k-offset (m*NS + f*64) added per step
    const float* aptr = xs_base + (size_t)n * FB + bBase + laneM;

    int colBase = colGrp * 64;
    int colLane = colBase + laneM;

    // incremental K walk: k = f*5 + m, offA = m*NS + f*64, offW = k*OC
    int m1 = koff;     int offA1 = koff * NSi;       int offW1 = koff * OC;
    int m2 = koff + 1; int offA2 = (koff + 1) * NSi; int offW2 = (koff + 1) * OC;

    v8f acc[2][4] = {};

    for (int kk = 0; kk < 82; ++kk) {
        v2f a0, a1;
        a0.x = aptr[offA1];      a0.y = aptr[offA2];
        a1.x = aptr[offA1 + 16]; a1.y = aptr[offA2 + 16];
#pragma unroll
        for (int s = 0; s < 4; ++s) {
            int cW = colLane + s * 16;
            v2f bf;
            bf.x = W[offW1 + cW];
            bf.y = W[offW2 + cW];
            acc[0][s] = __builtin_amdgcn_wmma_f32_16x16x4_f32(
                            false, a0, false, bf, (short)0, acc[0][s], false, false);
            acc[1][s] = __builtin_amdgcn_wmma_f32_16x16x4_f32(
                            false, a1, false, bf, (short)0, acc[1][s], false, false);
        }
        // k += 4:  m' = m+4 (wrap: m-1, f+1) ; wrap delta = 64 - NS, else 4*NS
        m1 += 4; bool w1 = (m1 >= 5); offA1 += w1 ? (B_BATCH - NSi) : 4 * NSi; if (w1) m1 -= 5;
        m2 += 4; bool w2 = (m2 >= 5); offA2 += w2 ? (B_BATCH - NSi) : 4 * NSi; if (w2) m2 -= 5;
        offW1 += 4 * OC; offW2 += 4 * OC;
    }

    // tail: k base 328; lo half lanes cover k=328,329 (valid), hi half k=330,331 -> 0
    {
        bool valid = (hi == 0);
        v2f a0, a1;
        a0.x = valid ? aptr[offA1]      : 0.f;
        a0.y = valid ? aptr[offA2]      : 0.f;
        a1.x = valid ? aptr[offA1 + 16] : 0.f;
        a1.y = valid ? aptr[offA2 + 16] : 0.f;
#pragma unroll
        for (int s = 0; s < 4; ++s) {
            int cW = colLane + s * 16;
            v2f bf;
            bf.x = valid ? W[offW1 + cW] : 0.f;
            bf.y = valid ? W[offW2 + cW] : 0.f;
            acc[0][s] = __builtin_amdgcn_wmma_f32_16x16x4_f32(
                            false, a0, false, bf, (short)0, acc[0][s], false, false);
            acc[1][s] = __builtin_amdgcn_wmma_f32_16x16x4_f32(
                            false, a1, false, bf, (short)0, acc[1][s], false, false);
        }
    }

    // epilogue: D layout — lane gives N (col), VGPR v gives M (row = v + hi*8)
#pragma unroll
    for (int t = 0; t < 2; ++t) {
#pragma unroll
        for (int s = 0; s < 4; ++s) {
            int cc = colBase + s * 16 + laneM;
            float bia = bias[cc];
#pragma unroll
            for (int v = 0; v < 8; ++v) {
                int brow = bBase + t * 16 + v + hi * 8;
                size_t row = (size_t)brow * N_NODES + n;
                float val = acc[t][s][v] + bia;
                if (!FUSE_FINAL) {
                    // r/u gates: sigmoid, write value[row, 128]
                    out[row * OC + cc] = 1.f / (1.f + __expf(-val));
                } else {
                    // c = tanh(.); out = u*hx + (1-u)*c
                    float c = tanhf(val);
                    float u = value[row * 128 + 64 + cc];
                    float h = hx[row * 64 + cc];
                    out[row * 64 + cc] = u * h + (1.f - u) * c;
                }
            }
        }
    }
}

// ---------------- host ----------------
extern "C" void kernel_launch(void* const* d_in, const int* in_sizes, int n_in,
                              void* d_out, int out_size, void* d_ws, size_t ws_size,
                              hipStream_t stream)
{
    const float* inputs   = (const float*)d_in[0];
    const float* hx       = (const float*)d_in[1];
    const int*   sup_rows = (const int*)  d_in[2];
    const int*   sup_cols = (const int*)  d_in[3];
    const float* sup_vals = (const float*)d_in[4];
    const float* W_ru     = (const float*)d_in[5];
    const float* b_ru     = (const float*)d_in[6];
    const float* W_c      = (const float*)d_in[7];
    const float* b_c      = (const float*)d_in[8];
    float* out = (float*)d_out;

    float* ws    = (float*)d_ws;
    float* xs    = ws;              // 5 * NS floats (xcat buffers m = 0..4)
    float* value = ws + 5 * NS;     // R_ROWS * 128 floats (sigmoid(r|u))

    const unsigned elemBlocks = (unsigned)((NS + 255) / 256);
    const unsigned zeroBlocks = (unsigned)((NS / 4 + 255) / 256);
    const dim3 spmmGrid(NNZ_E, (FB / 4 + 255) / 256);

    auto spmm_run = [&](int s, const float* x, float* y) {
        zero_kernel<<<zeroBlocks, 256, 0, stream>>>((float4*)y, NS / 4);
        spmm_kernel<<<spmmGrid, 256, 0, stream>>>(sup_rows + s * NNZ_E,
                                                  sup_cols + s * NNZ_E,
                                                  sup_vals + s * NNZ_E, x, y);
    };
    auto gconv_sparse = [&]() {   // fills xs[1..4] from xs[0]
        spmm_run(0, xs,          xs + 1 * NS);
        spmm_run(0, xs + 1 * NS, xs + 2 * NS);
        cheb_kernel<<<elemBlocks, 256, 0, stream>>>(xs + 2 * NS, xs);
        spmm_run(1, xs,          xs + 3 * NS);
        spmm_run(1, xs + 3 * NS, xs + 4 * NS);
        cheb_kernel<<<elemBlocks, 256, 0, stream>>>(xs + 4 * NS, xs);
    };

    // ---- gconv 1: gates r,u = sigmoid(gconv(inputs, hx) @ W_ru + 1) ----
    build_x0_kernel<<<elemBlocks, 256, 0, stream>>>(inputs, hx, (const float*)nullptr, xs, 0);
    gconv_sparse();
    // waves = N * 2 bGroups * 2 colGroups = 16384 -> 2048 blocks of 8 waves
    gemm_wmma_kernel<128, false><<<N_NODES * 2 * 2 / 8, 256, 0, stream>>>(
        xs, W_ru, b_ru, nullptr, nullptr, value);

    // ---- gconv 2: c = tanh(gconv(inputs, r*hx) @ W_c); out = u*hx + (1-u)*c ----
    build_x0_kernel<<<elemBlocks, 256, 0, stream>>>(inputs, hx, value, xs, 1);
    gconv_sparse();
    // waves = N * 2 bGroups * 1 colGroup = 8192 -> 1024 blocks of 8 waves
    gemm_wmma_kernel<64, true><<<N_NODES * 2 * 1 / 8, 256, 0, stream>>>(
        xs, W_c, b_c, value, hx, out);
}